// TripletLoss_81209241633403
// MI455X (gfx1250) — compile-verified
//
#include <hip/hip_runtime.h>

typedef __attribute__((ext_vector_type(2))) float v2f;
typedef __attribute__((ext_vector_type(8))) float v8f;

#define DIM      128
#define T_TRIP   500000
#define MARGIN_F 1.0f
#define EPS_F    1e-6f
#define GROUPS   (T_TRIP / 16)   // 31250, exact

__global__ void tl_init_acc(double* acc) {
    if (threadIdx.x == 0 && blockIdx.x == 0) *acc = 0.0;
}

// One wave handles 16 triplets. Lane l owns triplet (group*16 + (l&15)).
// V_WMMA_F32_16X16X4_F32 operand layout (identical for A and B):
//   v[0] = {lanes 0-15: K0, lanes 16-31: K2}, v[1] = {K1 | K3}
// so lane l only needs the contiguous float2 at row + 4c + (l>=16 ? 2 : 0).
// Passing the diff vector as both A and B accumulates C += D*D^T, whose
// diagonal holds the 16 squared distances — no cross-lane ops in the loop.
__global__ __launch_bounds__(256) void tl_main(const float* __restrict__ batch,
                                               const int*   __restrict__ trip,
                                               double*      __restrict__ acc) {
    const int lane  = threadIdx.x & 31;
    const int group = (blockIdx.x * blockDim.x + threadIdx.x) >> 5;
    if (group >= GROUPS) return;                 // wave-uniform exit only

    const int row = lane & 15;
    const int t   = group * 16 + row;
    const int ia  = trip[t * 3 + 0];
    const int ipo = trip[t * 3 + 1];
    const int ine = trip[t * 3 + 2];

    const int sel = (lane >= 16) ? 2 : 0;        // low half: K0,K1; high: K2,K3
    const float* __restrict__ ar = batch + (size_t)ia  * DIM + sel;
    const float* __restrict__ pr = batch + (size_t)ipo * DIM + sel;
    const float* __restrict__ nr = batch + (size_t)ine * DIM + sel;

    const v2f eps2 = {EPS_F, EPS_F};
    v8f acc_ap = {};
    v8f acc_an = {};

#pragma unroll 4
    for (int c = 0; c < DIM / 4; ++c) {
        const v2f av = *(const v2f*)(ar + 4 * c);
        const v2f pv = *(const v2f*)(pr + 4 * c);
        const v2f nv = *(const v2f*)(nr + 4 * c);

        const v2f dap = av - pv + eps2;
        const v2f dan = av - nv + eps2;

        // 8 args: (neg_a, A, neg_b, B, c_mod, C, reuse_a, reuse_b)
        acc_ap = __builtin_amdgcn_wmma_f32_16x16x4_f32(
            false, dap, false, dap, (short)0, acc_ap, false, false);
        acc_an = __builtin_amdgcn_wmma_f32_16x16x4_f32(
            false, dan, false, dan, (short)0, acc_an, false, false);
    }

    // Diagonal of (Gram_ap - Gram_an): triplet m lives at
    //   m in 0..7 :  reg m,   lane m
    //   m in 8..15:  reg m-8, lane m+16  (i.e., lanes 24..31 hold m via reg lane-24)
    const v8f dv = acc_ap - acc_an;
    const bool valid = (lane < 8) || (lane >= 24);
    const int  r = valid ? ((lane < 8) ? lane : (lane - 24)) : 0;

    const float x0 = (r & 1) ? dv[1] : dv[0];
    const float x1 = (r & 1) ? dv[3] : dv[2];
    const float x2 = (r & 1) ? dv[5] : dv[4];
    const float x3 = (r & 1) ? dv[7] : dv[6];
    const float y0 = (r & 2) ? x1 : x0;
    const float y1 = (r & 2) ? x3 : x2;
    const float dg = (r & 4) ? y1 : y0;

    float loss = valid ? fmaxf(dg + MARGIN_F, 0.0f) : 0.0f;

    // wave32 reduction over the 16 valid diag-carrying lanes
#pragma unroll
    for (int off = 16; off >= 1; off >>= 1)
        loss += __shfl_xor(loss, off, 32);

    if (lane == 0)
        unsafeAtomicAdd(acc, (double)loss);      // global_atomic_add_f64
}

__global__ void tl_finalize(const double* __restrict__ acc,
                            float* __restrict__ out) {
    if (threadIdx.x == 0 && blockIdx.x == 0)
        out[0] = (float)(*acc / (double)T_TRIP);
}

extern "C" void kernel_launch(void* const* d_in, const int* in_sizes, int n_in,
                              void* d_out, int out_size, void* d_ws, size_t ws_size,
                              hipStream_t stream) {
    const float* batch = (const float*)d_in[0];
    const int*   trip  = (const int*)d_in[1];
    double*      acc   = (double*)d_ws;
    float*       out   = (float*)d_out;

    hipLaunchKernelGGL(tl_init_acc, dim3(1), dim3(1), 0, stream, acc);

    const int waves_per_block = 8;               // 256 threads, wave32
    const int blocks = (GROUPS + waves_per_block - 1) / waves_per_block;
    hipLaunchKernelGGL(tl_main, dim3(blocks), dim3(256), 0, stream,
                       batch, trip, acc);

    hipLaunchKernelGGL(tl_finalize, dim3(1), dim3(1), 0, stream, acc, out);
}